// HandoverGCN_77163382440451
// MI455X (gfx1250) — compile-verified
//
#include <hip/hip_runtime.h>
#include <hip/hip_bf16.h>
#include <math.h>

typedef __attribute__((ext_vector_type(2))) float v2f;
typedef __attribute__((ext_vector_type(8))) float v8f;

#define DFEAT 128
#define H1DIM 32
#define H2DIM 16
#define NCLS  8

// ---------------------------------------------------------------- degree ----
__global__ void k_init_deg(float* __restrict__ deg, int N) {
    int i = blockIdx.x * blockDim.x + threadIdx.x;
    if (i < N) deg[i] = 1.0f;                     // self-loop contributes 1
}

__global__ void k_acc_deg(const long long* __restrict__ ei, float* __restrict__ deg, int E) {
    int e = blockIdx.x * blockDim.x + threadIdx.x;
    if (e < E) {
        int d = (int)ei[(size_t)E + e];           // edge_index[1][e]
        atomicAdd(&deg[d], 1.0f);
    }
}

__global__ void k_rsqrt(float* __restrict__ deg, int N) {
    int i = blockIdx.x * blockDim.x + threadIdx.x;
    if (i < N) {
        float d = deg[i];
        deg[i] = (d > 0.0f) ? rsqrtf(d) : 0.0f;   // becomes dinv
    }
}

// ------------------------------------------------------- WMMA fp32 GEMMs ----
// h1[N,32] = x[N,128] @ W1[128,32]; one wave per 16x16 tile, K-loop of 4.
__global__ void k_gemm1_wmma(const float* __restrict__ X, const float* __restrict__ W,
                             float* __restrict__ H, int N) {
    int wave = (int)((blockIdx.x * blockDim.x + threadIdx.x) >> 5);
    int lane = threadIdx.x & 31;
    int mtiles = (N + 15) >> 4;
    if (wave >= mtiles * 2) return;               // wave-uniform: EXEC stays all-1
    int r0 = (wave >> 1) * 16;
    int c0 = (wave & 1) * 16;
    int m  = lane & 15;                           // row (A) / col (B,C)
    int kh = (lane >> 4) * 2;                     // lanes 16-31 hold K+2,K+3
    int rr = r0 + m; if (rr > N - 1) rr = N - 1;  // clamp loads for partial tile
    const float* xrow = X + (size_t)rr * DFEAT;

    v8f acc = {};
#pragma unroll 8
    for (int k = 0; k < DFEAT; k += 4) {
        int kk = k + kh;
        v2f a = *reinterpret_cast<const v2f*>(xrow + kk);      // 8B aligned
        v2f b;
        b.x = W[(size_t)kk       * H1DIM + c0 + m];
        b.y = W[(size_t)(kk + 1) * H1DIM + c0 + m];
        acc = __builtin_amdgcn_wmma_f32_16x16x4_f32(false, a, false, b,
                                                    (short)0, acc, false, false);
    }
    // C/D layout: VGPR v holds row r0 + v + 8*(lane>=16), col = lane&15
    int rb = (lane >> 4) * 8;
#pragma unroll
    for (int v = 0; v < 8; ++v) {
        int row = r0 + rb + v;
        if (row < N) H[(size_t)row * H1DIM + c0 + m] = acc[v];
    }
}

// h2[N,16] = relu(agg1)[N,32] @ W2[32,16]; ReLU fused into A-fragment load.
__global__ void k_gemm2_wmma(const float* __restrict__ A1, const float* __restrict__ W,
                             float* __restrict__ H2, int N) {
    int wave = (int)((blockIdx.x * blockDim.x + threadIdx.x) >> 5);
    int lane = threadIdx.x & 31;
    int mtiles = (N + 15) >> 4;
    if (wave >= mtiles) return;
    int r0 = wave * 16;
    int m  = lane & 15;
    int kh = (lane >> 4) * 2;
    int rr = r0 + m; if (rr > N - 1) rr = N - 1;
    const float* arow = A1 + (size_t)rr * H1DIM;

    v8f acc = {};
#pragma unroll
    for (int k = 0; k < H1DIM; k += 4) {
        int kk = k + kh;
        v2f a;
        a.x = fmaxf(arow[kk],     0.0f);
        a.y = fmaxf(arow[kk + 1], 0.0f);
        v2f b;
        b.x = W[(size_t)kk       * H2DIM + m];
        b.y = W[(size_t)(kk + 1) * H2DIM + m];
        acc = __builtin_amdgcn_wmma_f32_16x16x4_f32(false, a, false, b,
                                                    (short)0, acc, false, false);
    }
    int rb = (lane >> 4) * 8;
#pragma unroll
    for (int v = 0; v < 8; ++v) {
        int row = r0 + rb + v;
        if (row < N) H2[(size_t)row * H2DIM + m] = acc[v];
    }
}

// --------------------------------------------- aggregation init + scatter ---
// agg[i][c] = h[i][c]*dinv[i]^2 + bias[c]   (self-loop msg + bias, pre-scatter)
__global__ void k_init_agg(const float* __restrict__ h, const float* __restrict__ dinv,
                           const float* __restrict__ bias, float* __restrict__ agg,
                           int total, int cshift) {
    int t = blockIdx.x * blockDim.x + threadIdx.x;
    if (t >= total) return;
    int i = t >> cshift;
    int c = t & ((1 << cshift) - 1);
    float di = dinv[i];
    agg[t] = h[t] * (di * di) + bias[c];
}

// Layer-1 scatter: 4 threads/edge, 8 floats each (two float4 loads, 8 atomics)
__global__ void k_scatter32(const long long* __restrict__ ei, const float* __restrict__ h,
                            const float* __restrict__ dinv, float* __restrict__ agg, int E) {
    int t = blockIdx.x * blockDim.x + threadIdx.x;
    if (t >= E * 4) return;
    int e = t >> 2;
    int q = (t & 3) * 8;
    int s = (int)ei[e];
    int d = (int)ei[(size_t)E + e];
    float w = dinv[s] * dinv[d];
    const float4* src = reinterpret_cast<const float4*>(h + (size_t)s * H1DIM + q);
    float4 m0 = src[0];
    float4 m1 = src[1];
    float* out = agg + (size_t)d * H1DIM + q;
    atomicAdd(out + 0, m0.x * w); atomicAdd(out + 1, m0.y * w);
    atomicAdd(out + 2, m0.z * w); atomicAdd(out + 3, m0.w * w);
    atomicAdd(out + 4, m1.x * w); atomicAdd(out + 5, m1.y * w);
    atomicAdd(out + 6, m1.z * w); atomicAdd(out + 7, m1.w * w);
}

// Layer-2 scatter: 2 threads/edge, 8 floats each
__global__ void k_scatter16(const long long* __restrict__ ei, const float* __restrict__ h,
                            const float* __restrict__ dinv, float* __restrict__ agg, int E) {
    int t = blockIdx.x * blockDim.x + threadIdx.x;
    if (t >= E * 2) return;
    int e = t >> 1;
    int q = (t & 1) * 8;
    int s = (int)ei[e];
    int d = (int)ei[(size_t)E + e];
    float w = dinv[s] * dinv[d];
    const float4* src = reinterpret_cast<const float4*>(h + (size_t)s * H2DIM + q);
    float4 m0 = src[0];
    float4 m1 = src[1];
    float* out = agg + (size_t)d * H2DIM + q;
    atomicAdd(out + 0, m0.x * w); atomicAdd(out + 1, m0.y * w);
    atomicAdd(out + 2, m0.z * w); atomicAdd(out + 3, m0.w * w);
    atomicAdd(out + 4, m1.x * w); atomicAdd(out + 5, m1.y * w);
    atomicAdd(out + 6, m1.z * w); atomicAdd(out + 7, m1.w * w);
}

// ------------------------------- fused ReLU + classifier + log_softmax ------
__global__ void k_classify(const float* __restrict__ agg2, const float* __restrict__ Wc,
                           const float* __restrict__ bc, float* __restrict__ out, int N) {
    __shared__ float sW[H2DIM * NCLS];
    __shared__ float sB[NCLS];
    if (threadIdx.x < H2DIM * NCLS) sW[threadIdx.x] = Wc[threadIdx.x];
    if (threadIdx.x < NCLS)         sB[threadIdx.x] = bc[threadIdx.x];
    __syncthreads();
    int i = blockIdx.x * blockDim.x + threadIdx.x;
    if (i >= N) return;

    float v[H2DIM];
    const float4* row = reinterpret_cast<const float4*>(agg2 + (size_t)i * H2DIM);
#pragma unroll
    for (int c4 = 0; c4 < 4; ++c4) {
        float4 r = row[c4];
        v[c4 * 4 + 0] = fmaxf(r.x, 0.0f);
        v[c4 * 4 + 1] = fmaxf(r.y, 0.0f);
        v[c4 * 4 + 2] = fmaxf(r.z, 0.0f);
        v[c4 * 4 + 3] = fmaxf(r.w, 0.0f);
    }
    float l[NCLS];
#pragma unroll
    for (int k = 0; k < NCLS; ++k) {
        float s = sB[k];
#pragma unroll
        for (int c = 0; c < H2DIM; ++c) s = fmaf(v[c], sW[c * NCLS + k], s);
        l[k] = s;
    }
    float mx = l[0];
#pragma unroll
    for (int k = 1; k < NCLS; ++k) mx = fmaxf(mx, l[k]);
    float se = 0.0f;
#pragma unroll
    for (int k = 0; k < NCLS; ++k) se += expf(l[k] - mx);
    float lse = logf(se);
    float* o = out + (size_t)i * NCLS;
#pragma unroll
    for (int k = 0; k < NCLS; ++k) o[k] = l[k] - mx - lse;
}

// ----------------------------------------------------------------- launch ---
extern "C" void kernel_launch(void* const* d_in, const int* in_sizes, int n_in,
                              void* d_out, int out_size, void* d_ws, size_t ws_size,
                              hipStream_t stream) {
    const float*     x  = (const float*)d_in[0];
    const long long* ei = (const long long*)d_in[1];   // int64 edge_index [2,E]
    const float*     W1 = (const float*)d_in[2];
    const float*     b1 = (const float*)d_in[3];
    const float*     W2 = (const float*)d_in[4];
    const float*     b2 = (const float*)d_in[5];
    const float*     Wc = (const float*)d_in[6];
    const float*     bc = (const float*)d_in[7];
    float*           out = (float*)d_out;

    int N = in_sizes[0] / DFEAT;     // 100000
    int E = in_sizes[1] / 2;         // 3200000

    // carve workspace (256B aligned regions)
    char*  ws  = (char*)d_ws;
    size_t off = 0;
    auto carve = [&](size_t bytes) -> void* {
        void* p = ws + off;
        off += (bytes + 255) & ~(size_t)255;
        return p;
    };
    float* dinv = (float*)carve((size_t)N * sizeof(float));
    float* h1   = (float*)carve((size_t)N * H1DIM * sizeof(float));
    float* agg1 = (float*)carve((size_t)N * H1DIM * sizeof(float));
    float* h2   = (float*)carve((size_t)N * H2DIM * sizeof(float));
    float* agg2 = (float*)carve((size_t)N * H2DIM * sizeof(float));

    const int B = 256;
    auto blk = [](long long n, int b) { return (int)((n + b - 1) / b); };

    // degrees -> dinv
    k_init_deg<<<blk(N, B), B, 0, stream>>>(dinv, N);
    k_acc_deg <<<blk(E, B), B, 0, stream>>>(ei, dinv, E);
    k_rsqrt   <<<blk(N, B), B, 0, stream>>>(dinv, N);

    // layer 1: WMMA GEMM, init (bias + self-loop), edge scatter
    int mt = (N + 15) >> 4;
    k_gemm1_wmma<<<blk((long long)mt * 2 * 32, B), B, 0, stream>>>(x, W1, h1, N);
    k_init_agg  <<<blk((long long)N * H1DIM, B), B, 0, stream>>>(h1, dinv, b1, agg1, N * H1DIM, 5);
    k_scatter32 <<<blk((long long)E * 4, B), B, 0, stream>>>(ei, h1, dinv, agg1, E);

    // layer 2 (ReLU fused into GEMM2 A-load)
    k_gemm2_wmma<<<blk((long long)mt * 32, B), B, 0, stream>>>(agg1, W2, h2, N);
    k_init_agg  <<<blk((long long)N * H2DIM, B), B, 0, stream>>>(h2, dinv, b2, agg2, N * H2DIM, 4);
    k_scatter16 <<<blk((long long)E * 2, B), B, 0, stream>>>(ei, h2, dinv, agg2, E);

    // classifier + log_softmax (ReLU fused)
    k_classify<<<blk(N, B), B, 0, stream>>>(agg2, Wc, bc, out, N);
}